// Step1Model_34840774705369
// MI455X (gfx1250) — compile-verified
//
#include <hip/hip_runtime.h>
#include <hip/hip_bf16.h>
#include <math.h>

typedef float v2f __attribute__((ext_vector_type(2)));
typedef float v8f __attribute__((ext_vector_type(8)));
typedef unsigned int v4u __attribute__((ext_vector_type(4)));
typedef int v4i __attribute__((ext_vector_type(4)));
typedef int v8i __attribute__((ext_vector_type(8)));

#define V_   8
#define T_   4096
#define DIN_ 512
#define D_   1024
#define DA_  1024
#define KR_  16

// ---------------------------------------------------------------------------
// TDM: DMA a 2D fp32 tile (tile_x contiguous elems x tile_y rows, row stride
// `stride` elems) from global into LDS at byte offset `lds_off`, inserting
// 1 pad DWORD after every 16 DWORDs -> LDS row stride 17 floats.
// Descriptor layout per CDNA5 ISA ch.8 (async tensor). 6-arg builtin
// (clang-23 toolchain): extra v8i group before cpol, zero-filled.
// ---------------------------------------------------------------------------
__device__ __forceinline__ void tdm_load_tile17(unsigned lds_off, const float* gptr,
                                                int tile_x, int tile_y, int stride) {
  unsigned long long ga = (unsigned long long)(uintptr_t)gptr;
  v4u g0;
  g0.x = 1u;                                              // count=1, user desc
  g0.y = lds_off;                                         // lds_addr (bytes)
  g0.z = (unsigned)(ga & 0xFFFFFFFFu);                    // global_addr[31:0]
  g0.w = (unsigned)((ga >> 32) & 0x01FFFFFFu) | (2u << 30); // addr[56:32] | type=2
  const unsigned td0 = 0x7FFFFFFFu, td1 = 0x7FFFFFFFu;    // huge tensor dims (no OOB)
  v8i g1;
  g1[0] = (int)((2u << 16)      // data_size = 4B
              | (1u << 20)      // pad_enable
              | (3u << 22)      // pad_interval: 16 DWORDs
              | (0u << 25));    // pad_amount: 1 DWORD
  g1[1] = (int)((td0 & 0xFFFFu) << 16);
  g1[2] = (int)(((td0 >> 16) & 0xFFFFu) | ((td1 & 0xFFFFu) << 16));
  g1[3] = (int)(((td1 >> 16) & 0xFFFFu) | ((unsigned)tile_x << 16));  // tile_dim0
  g1[4] = (int)((unsigned)tile_y);                        // tile_dim1 (tile_dim2=0)
  g1[5] = (int)(unsigned)stride;                          // tensor_dim0_stride lo32
  g1[6] = 0;
  g1[7] = 0;
  v4i gz4 = {};
  v8i gz8 = {};
  __builtin_amdgcn_tensor_load_to_lds(g0, g1, gz4, gz4, gz8, 0);
}

// ---------------------------------------------------------------------------
// Tiled fp32 WMMA GEMM:  C[M,N] = act(A[M,K] * B(+bias))
//   TRANSB=false: B row-major [K,N] (manual transpose-staging, double buffered)
//   TRANSB=true : B row-major [N,K] (TDM-staged like A)
//   GATED: if *gate==0, C tile = A tile (requires K==N)
// Block 256 threads (8 waves); 128x128 tile, BK=16, double-buffered LDS.
// Wave tiling 2x4, each wave 64x32 -> 4x2 v8f accumulators.
// Both LDS tiles stored element-major [row-of-16-frag][k] stride 17 so WMMA
// fragment pairs are adjacent dwords (ds_load_2addr offset1:1, no repack movs).
// ---------------------------------------------------------------------------
template <bool RELU, bool TRANSB, bool GATED>
__global__ __launch_bounds__(256) void gemm_wmma_f32(
    const float* __restrict__ A, const float* __restrict__ B,
    const float* __restrict__ bias, float* __restrict__ C,
    int M, int N, int K,
    long sA, long sB, long sBias, long sC,
    const int* __restrict__ gate)
{
  const int v = blockIdx.z;
  A += (long)v * sA;
  B += (long)v * sB;
  C += (long)v * sC;

  const int tid = threadIdx.x;
  const int bm  = blockIdx.x * 128;
  const int bn  = blockIdx.y * 128;

  if (GATED) {
    if (*gate == 0) {   // uniform across the grid
      for (int i = tid; i < 128 * 128; i += 256) {
        int r = i >> 7, c = i & 127;
        C[(long)(bm + r) * N + (bn + c)] = A[(long)(bm + r) * K + (bn + c)];
      }
      return;
    }
  }

  __shared__ float As[2][128][17];   // [m][k]
  __shared__ float Bs[2][128][17];   // [n][k]

  const int lane = tid & 31;
  const int hh   = lane >> 4;        // lane half
  const int lr   = lane & 15;
  const int wid  = tid >> 5;
  const int wm   = (wid >> 2) * 64;  // wave M offset
  const int wn   = (wid & 3) * 32;   // wave N offset

  const unsigned ldsA[2] = {(unsigned)(uintptr_t)&As[0][0][0],
                            (unsigned)(uintptr_t)&As[1][0][0]};
  const unsigned ldsB[2] = {(unsigned)(uintptr_t)&Bs[0][0][0],
                            (unsigned)(uintptr_t)&Bs[1][0][0]};

  v8f acc[4][2] = {};
  float4 breg0, breg1;

  auto issue_tiles = [&](int kt, int buf) {
    if (tid < 32) {                  // one wave issues the DMAs
      tdm_load_tile17(ldsA[buf], A + (long)bm * K + kt, 16, 128, K);
      if (TRANSB)
        tdm_load_tile17(ldsB[buf], B + (long)bn * K + kt, 16, 128, K);
    }
  };
  auto loadB_regs = [&](int kt) {    // !TRANSB: B[K,N] rows -> registers
    int i0 = tid, i1 = tid + 256;
    breg0 = *(const float4*)&B[(long)(kt + (i0 >> 5)) * N + bn + ((i0 & 31) << 2)];
    breg1 = *(const float4*)&B[(long)(kt + (i1 >> 5)) * N + bn + ((i1 & 31) << 2)];
  };
  auto storeB_lds = [&](int buf) {   // scatter-transpose into Bs[n][k]
    int i0 = tid, r0 = i0 >> 5, c0 = (i0 & 31) << 2;
    Bs[buf][c0 + 0][r0] = breg0.x; Bs[buf][c0 + 1][r0] = breg0.y;
    Bs[buf][c0 + 2][r0] = breg0.z; Bs[buf][c0 + 3][r0] = breg0.w;
    int i1 = tid + 256, r1 = i1 >> 5, c1 = (i1 & 31) << 2;
    Bs[buf][c1 + 0][r1] = breg1.x; Bs[buf][c1 + 1][r1] = breg1.y;
    Bs[buf][c1 + 2][r1] = breg1.z; Bs[buf][c1 + 3][r1] = breg1.w;
  };

  // prologue: stage tile 0 into buffer 0
  issue_tiles(0, 0);
  if (!TRANSB) { loadB_regs(0); storeB_lds(0); }
  if (tid < 32) __builtin_amdgcn_s_wait_tensorcnt(0);
  __syncthreads();

  const int nkt = K >> 4;
  for (int t = 0; t < nkt; ++t) {
    const int cur = t & 1, nxt = cur ^ 1;
    const bool more = (t + 1 < nkt);
    if (more) {
      issue_tiles((t + 1) << 4, nxt);         // DMA next tile during compute
      if (!TRANSB) loadB_regs((t + 1) << 4);  // global loads in flight
    }

#pragma unroll
    for (int k0 = 0; k0 < 16; k0 += 4) {
      v2f a[4], b[2];
#pragma unroll
      for (int i = 0; i < 4; ++i) {
        int r = wm + i * 16 + lr;            // A(M,K): lane=(K/2)*16+M, vgpr=K%2
        a[i].x = As[cur][r][k0 + 2 * hh + 0];
        a[i].y = As[cur][r][k0 + 2 * hh + 1];
      }
#pragma unroll
      for (int j = 0; j < 2; ++j) {
        int c = wn + j * 16 + lr;            // B(K,N): lane=(K/2)*16+N, vgpr=K%2
        b[j].x = Bs[cur][c][k0 + 2 * hh + 0];
        b[j].y = Bs[cur][c][k0 + 2 * hh + 1];
      }
#pragma unroll
      for (int i = 0; i < 4; ++i)
#pragma unroll
        for (int j = 0; j < 2; ++j)
          acc[i][j] = __builtin_amdgcn_wmma_f32_16x16x4_f32(
              false, a[i], false, b[j], (short)0, acc[i][j], false, false);
    }

    if (more && !TRANSB) storeB_lds(nxt);
    if (more && tid < 32) __builtin_amdgcn_s_wait_tensorcnt(0);
    __syncthreads();
  }

  // epilogue: D(m,n): vgpr=m%8, lane=(m/8)*16+n
  const float* biasv = bias ? bias + (long)v * sBias : nullptr;
#pragma unroll
  for (int i = 0; i < 4; ++i) {
#pragma unroll
    for (int j = 0; j < 2; ++j) {
      int col = bn + wn + j * 16 + lr;
      float bv = biasv ? biasv[col] : 0.0f;
#pragma unroll
      for (int r = 0; r < 8; ++r) {
        int row = bm + wm + i * 16 + r + hh * 8;
        float val = acc[i][j][r] + bv;
        if (RELU) val = fmaxf(val, 0.0f);
        C[(long)row * N + col] = val;
      }
    }
  }
}

// ---------------------------------------------------------------------------
// Small kernels
// ---------------------------------------------------------------------------
__global__ void activity_kernel(const float* __restrict__ x, float* __restrict__ act) {
  __shared__ float red[256];
  const int v = blockIdx.x;
  const long n = (long)T_ * DIN_;
  const float* xv = x + (long)v * n;
  float s = 0.f;
  for (long i = threadIdx.x; i < n; i += 256) s += fabsf(xv[i]);
  red[threadIdx.x] = s; __syncthreads();
  for (int st = 128; st > 0; st >>= 1) {
    if (threadIdx.x < st) red[threadIdx.x] += red[threadIdx.x + st];
    __syncthreads();
  }
  if (threadIdx.x == 0) act[v] = red[0] / (float)n;
}

__global__ void alpha_kernel(const float* __restrict__ act,
                             const float* __restrict__ omega_logits,
                             float* __restrict__ alpha) {
  if (threadIdx.x != 0) return;
  const float coefs[5] = {0.5f, 0.3f, 0.2f, 0.1f, 0.4f};
  float om[5], mx = -1e30f, sum = 0.f;
  for (int i = 0; i < 5; ++i) mx = fmaxf(mx, omega_logits[i]);
  for (int i = 0; i < 5; ++i) { om[i] = expf(omega_logits[i] - mx); sum += om[i]; }
  for (int i = 0; i < 5; ++i) om[i] /= sum;
  float Q[V_];
  for (int v = 0; v < V_; ++v) {
    float q = 0.f;
    for (int i = 0; i < 5; ++i)
      q += om[i] * (1.0f / (1.0f + expf(-coefs[i] * act[v])));
    Q[v] = q / 0.5f;  // / TAU_Q
  }
  mx = -1e30f; sum = 0.f;
  for (int v = 0; v < V_; ++v) mx = fmaxf(mx, Q[v]);
  float e[V_];
  for (int v = 0; v < V_; ++v) { e[v] = expf(Q[v] - mx); sum += e[v]; }
  for (int v = 0; v < V_; ++v) alpha[v] = e[v] / sum;
}

// out[v,c] = mean_t X[v,t,c]
__global__ void colmean_kernel(const float* __restrict__ X, float* __restrict__ out,
                               int rows, int cols) {
  const int v = blockIdx.y;
  const int c = blockIdx.x * 256 + threadIdx.x;
  const float* Xv = X + (long)v * rows * cols;
  float s = 0.f;
  for (int t = 0; t < rows; ++t) s += Xv[(long)t * cols + c];
  out[(long)v * cols + c] = s / (float)rows;
}

__global__ void router_kernel(const float* __restrict__ rW, const float* __restrict__ rb,
                              const float* __restrict__ alpha,
                              const float* __restrict__ g_view,
                              float* __restrict__ logits) {
  __shared__ float red[256];
  const int k = blockIdx.x;
  float s = 0.f;
  for (int j = threadIdx.x; j < V_ * D_; j += 256)
    s += rW[(long)k * (V_ * D_) + j] * alpha[j >> 10] * g_view[j];
  red[threadIdx.x] = s; __syncthreads();
  for (int st = 128; st > 0; st >>= 1) {
    if (threadIdx.x < st) red[threadIdx.x] += red[threadIdx.x + st];
    __syncthreads();
  }
  if (threadIdx.x == 0) logits[k] = red[0] + rb[k];
}

__global__ void pi_kernel(const float* __restrict__ logits, float* __restrict__ pi) {
  if (threadIdx.x != 0) return;
  float mx = -1e30f, sum = 0.f, e[KR_];
  for (int k = 0; k < KR_; ++k) mx = fmaxf(mx, logits[k]);
  for (int k = 0; k < KR_; ++k) { e[k] = expf(logits[k] - mx); sum += e[k]; }
  for (int k = 0; k < KR_; ++k) pi[k] = e[k] / sum;
}

__global__ void bx_kernel(const float* __restrict__ B_basis, const float* __restrict__ pi,
                          float* __restrict__ Bx) {
  const long i = (long)blockIdx.x * 256 + threadIdx.x;
  float s = 0.f;
  for (int k = 0; k < KR_; ++k) s += pi[k] * B_basis[(long)k * DA_ * D_ + i];
  Bx[i] = s;
}

__global__ void corr_kernel(const float* __restrict__ alpha, const float* __restrict__ z_view,
                            float* __restrict__ z_w, float* __restrict__ corr_out,
                            float* __restrict__ rho_out, int* __restrict__ gate) {
  __shared__ float Cm[V_][DA_];
  __shared__ float red[256];
  __shared__ float stat[V_];
  __shared__ float nrm[V_];
  __shared__ float cm[V_ * V_];
  const int tid = threadIdx.x;

  for (int i = tid; i < V_ * DA_; i += 256) {
    int v = i >> 10;
    float val = alpha[v] * z_view[i];
    z_w[i] = val;
    Cm[v][i & (DA_ - 1)] = val;
  }
  __syncthreads();
  for (int v = 0; v < V_; ++v) {  // row means
    float s = 0.f;
    for (int a = tid; a < DA_; a += 256) s += Cm[v][a];
    red[tid] = s; __syncthreads();
    for (int st = 128; st > 0; st >>= 1) {
      if (tid < st) red[tid] += red[tid + st];
      __syncthreads();
    }
    if (tid == 0) stat[v] = red[0] / (float)DA_;
    __syncthreads();
  }
  for (int i = tid; i < V_ * DA_; i += 256) Cm[i >> 10][i & (DA_ - 1)] -= stat[i >> 10];
  __syncthreads();
  for (int v = 0; v < V_; ++v) {  // norms
    float s = 0.f;
    for (int a = tid; a < DA_; a += 256) { float c = Cm[v][a]; s += c * c; }
    red[tid] = s; __syncthreads();
    for (int st = 128; st > 0; st >>= 1) {
      if (tid < st) red[tid] += red[tid + st];
      __syncthreads();
    }
    if (tid == 0) nrm[v] = sqrtf(red[0]);
    __syncthreads();
  }
  for (int p = 0; p < V_ * V_; ++p) {  // pairwise dots
    int u = p >> 3, w = p & 7;
    float s = 0.f;
    for (int a = tid; a < DA_; a += 256) s += Cm[u][a] * Cm[w][a];
    red[tid] = s; __syncthreads();
    for (int st = 128; st > 0; st >>= 1) {
      if (tid < st) red[tid] += red[tid + st];
      __syncthreads();
    }
    if (tid == 0) cm[p] = red[0];
    __syncthreads();
  }
  if (tid == 0) {
    float rho = -1e9f;
    for (int p = 0; p < V_ * V_; ++p) {
      int u = p >> 3, w = p & 7;
      float denom = fmaxf(nrm[u] * nrm[w], 1e-8f);
      float cf = cm[p] / denom;
      if (u == w) corr_out[p] = 0.0f;
      else { corr_out[p] = cf; rho = fmaxf(rho, cf); }
    }
    rho_out[0] = rho;
    gate[0] = (rho >= -1.0f) ? 1 : 0;
  }
}

// qkv[m][v][a] = sum_d z_w[v][d] * W_m[a][d]   (q = z_w @ Wq^T etc.)
__global__ void qkv_kernel(const float* __restrict__ z_w, const float* __restrict__ Wq,
                           const float* __restrict__ Wk, const float* __restrict__ Wv,
                           float* __restrict__ qkv) {
  const int m = blockIdx.y;
  const float* W = (m == 0) ? Wq : (m == 1) ? Wk : Wv;
  const int a = blockIdx.x * 256 + threadIdx.x;
  float acc[V_] = {};
  for (int d = 0; d < DA_; ++d) {
    float w = W[(long)a * DA_ + d];
    for (int v = 0; v < V_; ++v) acc[v] += z_w[v * DA_ + d] * w;
  }
  for (int v = 0; v < V_; ++v) qkv[((long)m * V_ + v) * DA_ + a] = acc[v];
}

__global__ void attn_z_kernel(const float* __restrict__ alpha, const float* __restrict__ z_view,
                              const float* __restrict__ qkv, const int* __restrict__ gate,
                              float* __restrict__ Z, float* __restrict__ cvec) {
  __shared__ float attn[V_][V_];
  const int tid = threadIdx.x;
  const float* q  = qkv;
  const float* k  = qkv + V_ * DA_;
  const float* vv = qkv + 2 * V_ * DA_;
  const bool g = (*gate != 0);
  if (g) {
    if (tid < V_ * V_) {
      int u = tid >> 3, w = tid & 7;
      float s = 0.f;
      for (int d = 0; d < DA_; ++d) s += q[u * DA_ + d] * k[w * DA_ + d];
      attn[u][w] = s * (1.0f / 32.0f);  // / sqrt(DA)
    }
    __syncthreads();
    if (tid < V_) {  // softmax row with -inf diagonal
      float mx = -1e30f;
      for (int w = 0; w < V_; ++w) if (w != tid) mx = fmaxf(mx, attn[tid][w]);
      float e[V_], sum = 0.f;
      for (int w = 0; w < V_; ++w) {
        e[w] = (w == tid) ? 0.0f : expf(attn[tid][w] - mx);
        sum += e[w];
      }
      for (int w = 0; w < V_; ++w) attn[tid][w] = e[w] / sum;
    }
    __syncthreads();
    if (tid < V_) {
      float s = 0.f;
      for (int v = 0; v < V_; ++v) s += alpha[v] * attn[v][tid];
      cvec[tid] = alpha[tid] * s;
    }
    for (int a = tid; a < DA_; a += 256) {
      float z = 0.f;
      for (int v = 0; v < V_; ++v) {
        float zp = 0.f;
        for (int u = 0; u < V_; ++u) zp += attn[v][u] * vv[u * DA_ + a];
        z += alpha[v] * zp;
      }
      Z[a] = z;
    }
  } else {
    if (tid < V_) cvec[tid] = alpha[tid];
    for (int a = tid; a < DA_; a += 256) {
      float z = 0.f;
      for (int v = 0; v < V_; ++v) z += alpha[v] * z_view[v * DA_ + a];
      Z[a] = z;
    }
  }
}

// Mbuf[t,a] = sum_u c[u] * h_aligned[u,t,a]
__global__ void combine_kernel(const float* __restrict__ h_al, const float* __restrict__ c,
                               float* __restrict__ Mb) {
  const long i = (long)blockIdx.x * 256 + threadIdx.x;
  float s = 0.f;
  for (int u = 0; u < V_; ++u) s += c[u] * h_al[(long)u * T_ * DA_ + i];
  Mb[i] = s;
}

// ---------------------------------------------------------------------------
extern "C" void kernel_launch(void* const* d_in, const int* in_sizes, int n_in,
                              void* d_out, int out_size, void* d_ws, size_t ws_size,
                              hipStream_t stream) {
  const float* x_wins   = (const float*)d_in[0];
  const float* W1       = (const float*)d_in[1];
  const float* b1       = (const float*)d_in[2];
  const float* W2       = (const float*)d_in[3];
  const float* b2       = (const float*)d_in[4];
  const float* router_W = (const float*)d_in[5];
  const float* router_b = (const float*)d_in[6];
  const float* B_basis  = (const float*)d_in[7];
  const float* Wq       = (const float*)d_in[8];
  const float* Wk       = (const float*)d_in[9];
  const float* Wv       = (const float*)d_in[10];
  const float* omega_l  = (const float*)d_in[11];

  float* out = (float*)d_out;
  // output layout: Z | H | h_aligned | alpha | pi | Bx | corr | rho
  const long OFF_Z     = 0;
  const long OFF_H     = OFF_Z + DA_;
  const long OFF_HAL   = OFF_H + (long)T_ * DA_;
  const long OFF_ALPHA = OFF_HAL + (long)V_ * T_ * DA_;
  const long OFF_PI    = OFF_ALPHA + V_;
  const long OFF_BX    = OFF_PI + KR_;
  const long OFF_CORR  = OFF_BX + (long)DA_ * D_;
  const long OFF_RHO   = OFF_CORR + V_ * V_;

  float* alpha = out + OFF_ALPHA;
  float* pi    = out + OFF_PI;
  float* Bx    = out + OFF_BX;
  float* h_al  = out + OFF_HAL;

  // workspace layout
  float* ws     = (float*)d_ws;
  float* h      = ws;                               // V*T*D
  float* h_raw  = ws + (long)V_ * T_ * D_;          // V*T*D
  float* Mbuf   = h;                                // alias: h dead after GEMM2
  float* sm     = ws + 2L * V_ * T_ * D_;
  float* act    = sm;            // 8
  float* g_view = sm + 8;        // 8192
  float* logits = sm + 8200;     // 16
  float* z_view = sm + 8216;     // 8192
  float* z_w    = sm + 16408;    // 8192
  float* qkv    = sm + 24600;    // 3*8*1024
  float* cvec   = sm + 49176;    // 8
  int*   gate   = (int*)(sm + 49184);

  // 1. quality gating -> alpha
  activity_kernel<<<V_, 256, 0, stream>>>(x_wins, act);
  alpha_kernel<<<1, 32, 0, stream>>>(act, omega_l, alpha);

  // 2. h = relu(x @ W1 + b1)
  gemm_wmma_f32<true, false, false><<<dim3(T_ / 128, D_ / 128, V_), 256, 0, stream>>>(
      x_wins, W1, b1, h, T_, D_, DIN_,
      (long)T_ * DIN_, (long)DIN_ * D_, (long)D_, (long)T_ * D_, nullptr);

  // 3. h_raw = h @ W2 + b2
  gemm_wmma_f32<false, false, false><<<dim3(T_ / 128, D_ / 128, V_), 256, 0, stream>>>(
      h, W2, b2, h_raw, T_, D_, D_,
      (long)T_ * D_, (long)D_ * D_, (long)D_, (long)T_ * D_, nullptr);

  // 4. routing: g_view = mean_t h_raw; pi = softmax(rW @ (alpha*g_view) + rb)
  colmean_kernel<<<dim3(D_ / 256, V_), 256, 0, stream>>>(h_raw, g_view, T_, D_);
  router_kernel<<<KR_, 256, 0, stream>>>(router_W, router_b, alpha, g_view, logits);
  pi_kernel<<<1, 32, 0, stream>>>(logits, pi);

  // 5. Bx = sum_k pi[k] * B_basis[k]
  bx_kernel<<<(DA_ * D_) / 256, 256, 0, stream>>>(B_basis, pi, Bx);

  // 6. h_aligned = h_raw @ Bx^T   (einsum 'ad,vtd->vta')
  gemm_wmma_f32<false, true, false><<<dim3(T_ / 128, DA_ / 128, V_), 256, 0, stream>>>(
      h_raw, Bx, nullptr, h_al, T_, DA_, D_,
      (long)T_ * D_, 0L, 0L, (long)T_ * DA_, nullptr);

  // 7. z_view = mean_t h_aligned; correlation gate
  colmean_kernel<<<dim3(DA_ / 256, V_), 256, 0, stream>>>(h_al, z_view, T_, DA_);
  corr_kernel<<<1, 256, 0, stream>>>(alpha, z_view, z_w, out + OFF_CORR, out + OFF_RHO, gate);

  // 8. attention head: q/k/v, attn, Z, combine weights c[u]
  qkv_kernel<<<dim3(DA_ / 256, 3), 256, 0, stream>>>(z_w, Wq, Wk, Wv, qkv);
  attn_z_kernel<<<1, 256, 0, stream>>>(alpha, z_view, qkv, gate, out + OFF_Z, cvec);

  // 9. M[t,a] = sum_u c[u]*h_aligned[u,t,a]; then H = gate ? M @ Wv^T : M
  combine_kernel<<<((long)T_ * DA_) / 256, 256, 0, stream>>>(h_al, cvec, Mbuf);
  gemm_wmma_f32<false, true, true><<<dim3(T_ / 128, DA_ / 128, 1), 256, 0, stream>>>(
      Mbuf, Wv, nullptr, out + OFF_H, T_, DA_, DA_,
      0L, 0L, 0L, 0L, gate);

  (void)in_sizes; (void)n_in; (void)out_size; (void)ws_size;
}